// SubtitleBranchLoss_56349970923626
// MI455X (gfx1250) — compile-verified
//
#include <hip/hip_runtime.h>
#include <stdint.h>

// ---------------------------------------------------------------------------
// SubtitleBranchLoss for MI455X (gfx1250).
// Memory-bound streaming reductions; CDNA5 Tensor Data Mover streams tiles
// into LDS, double-buffered with partial s_wait_tensorcnt waits (TDM ops from
// one wave complete in order, so tensorcnt<=3 means "previous buffer ready
// while the 3 prefetches stay in flight"). Hard-negative-mining top-k uses an
// exact MSB-first radix select on float bit patterns (losses >= 0, so the
// uint order matches float order); re-passes over the 26 MB scratch hit the
// 192 MB L2. All float accumulation is fixed-order tree reductions +
// per-block partial arrays (integer-only atomics) => deterministic.
// No WMMA: there is no matrix contraction in this loss; the bound is bytes
// (~131 MB unique => ~5.6 us floor at 23.3 TB/s), not math.
// ---------------------------------------------------------------------------

#ifndef __has_builtin
#define __has_builtin(x) 0
#endif
#if __has_builtin(__builtin_amdgcn_tensor_load_to_lds) && \
    __has_builtin(__builtin_amdgcn_s_wait_tensorcnt)
#define HAVE_TDM 1
#else
#define HAVE_TDM 0
#endif

typedef unsigned int v4u __attribute__((ext_vector_type(4)));
typedef int          v8i __attribute__((ext_vector_type(8)));
typedef int          v4i __attribute__((ext_vector_type(4)));

// ---- problem dimensions (fixed by setup_inputs) ----
#define NN 16
#define CC 32
#define HH 640
#define WW 640
#define HE 160
#define WE 160
#define S_TOT (NN * HH * WW)   // 6,553,600
#define EPIX  (HE * WE)        // 25,600
#define EPSF  1e-6f

// ---- tuning ----
#define NB1     800              // blocks for BCE pass 1 (4 chunks each)
#define NB2     1024             // blocks for sum-greater pass
#define NBH     1024             // blocks for histogram passes
#define CHUNK   2048             // floats per TDM tile in pass 1 (8 KB)
#define NCHUNK  (S_TOT / CHUNK)  // 3200
#define NCHE    5                // style chunks per (n,c)
#define CHUNK_E (EPIX / NCHE)    // 5120 floats (20 KB)

struct Acc {
  unsigned int posCnt, negCnt, kTotal, kNeed, prefix, countGreater, pad0, pad1;
  unsigned int hist[4][256];
  unsigned int npix[NN];
};

// ---- workspace layout (bytes); all float4-aligned ----
#define OFF_NEGLOSS ((size_t)0)
#define OFF_ACC     ((size_t)S_TOT * 4)                 // 26,214,400
#define OFF_POSPART (OFF_ACC + sizeof(Acc))             // Acc = 4192 B
#define OFF_NEGPART (OFF_POSPART + (size_t)NB1 * 4)
#define OFF_CHSUM   (OFF_NEGPART + (size_t)NB2 * 4)
#define OFF_CHSQ    (OFF_CHSUM + (size_t)NN * CC * 4)
#define OFF_MASKE   (OFF_CHSQ + (size_t)NN * CC * 4)    // NN*EPIX floats follow

// ---------------------------------------------------------------------------
// TDM: DMA a contiguous run of `nElems` f32 from global into LDS at byte
// offset `ldsByteOff`. Descriptor per CDNA5 ISA: a 1-row
// 2D tile (tile_dim0 = nElems <= 65535, tile_dim1 = 1, data_size = 4B).
// ---------------------------------------------------------------------------
#if HAVE_TDM
__device__ __forceinline__ void tdm_load_1d(const void* gptr,
                                            unsigned int ldsByteOff,
                                            unsigned int nElems) {
  uint64_t ga = (uint64_t)(uintptr_t)gptr;
  v4u g0;
  g0.x = 1u;                                        // count=1 (valid), user mode
  g0.y = ldsByteOff;                                // lds_addr [63:32]
  g0.z = (unsigned int)(ga & 0xFFFFFFFFu);          // global_addr[31:0]
  g0.w = (unsigned int)((ga >> 32) & 0x1FFFFFFu) | (2u << 30);  // addr[56:32], type=2
  v8i g1;
  g1[0] = (int)(2u << 16);                          // workgroup_mask=0, data_size=4B
  g1[1] = (int)((nElems & 0xFFFFu) << 16);          // tensor_dim0[15:0]
  g1[2] = (int)((nElems >> 16) & 0xFFFFu) | (1 << 16); // tensor_dim0[31:16], tensor_dim1=1
  g1[3] = (int)((nElems & 0xFFFFu) << 16);          // tile_dim0 = nElems
  g1[4] = 1;                                        // tile_dim1 = 1, tile_dim2 = 0
  g1[5] = (int)nElems;                              // tensor_dim0_stride[31:0]
  g1[6] = 0;                                        // stride hi, dim1_stride lo
  g1[7] = 0;
  v4i z4 = {0, 0, 0, 0};
#if defined(__clang_major__) && (__clang_major__ >= 23)
  v8i z8 = {0, 0, 0, 0, 0, 0, 0, 0};
  __builtin_amdgcn_tensor_load_to_lds(g0, g1, z4, z4, z8, 0);
#else
  __builtin_amdgcn_tensor_load_to_lds(g0, g1, z4, z4, 0);
#endif
}
#endif

// Fixed-order block reduce (blockDim == 256, wave32). Deterministic.
__device__ __forceinline__ float blockReduceF(float v, float* sbuf) {
#pragma unroll
  for (int o = 16; o > 0; o >>= 1) v += __shfl_down(v, o, 32);
  const int lane = (int)(threadIdx.x & 31u);
  const int wid  = (int)(threadIdx.x >> 5);
  if (lane == 0) sbuf[wid] = v;
  __syncthreads();
  v = (threadIdx.x < 8u) ? sbuf[threadIdx.x] : 0.0f;
  if (wid == 0) {
#pragma unroll
    for (int o = 4; o > 0; o >>= 1) v += __shfl_down(v, o, 32);
  }
  __syncthreads();
  return v;  // valid in thread 0
}

// ---------------------------------------------------------------------------
__global__ void k_init(Acc* acc) {
  unsigned int* p = (unsigned int*)acc;
  const int n = (int)(sizeof(Acc) / 4);
  for (int i = (int)threadIdx.x; i < n; i += (int)blockDim.x) p[i] = 0u;
}

// Elementwise BCE; nl = loss*neg (>= 0).
#define BCE1(P, G, M, OUT)                                        \
  {                                                               \
    float cp = fminf(fmaxf((P), 1e-12f), 1.0f);                   \
    float cq = fminf(fmaxf(1.0f - (P), 1e-12f), 1.0f);            \
    float ls = -((G) * logf(cp) + (1.0f - (G)) * logf(cq));       \
    if ((G) * (M) > 0.5f) { posSum += ls; pcnt += 1.0f; }         \
    bool ng = ((1.0f - (G)) * (M)) > 0.5f;                        \
    if (ng) ncnt += 1.0f;                                         \
    (OUT) = ng ? ls : 0.0f;                                       \
  }

// Pass 1: per-pixel BCE; write loss*neg to scratch; partial pos-sum/counts.
// TDM double-buffered: while computing buffer `buf`, wave 0 has already
// issued the next chunk's 3 DMAs into buf^1 and waited tensorcnt<=3.
__global__ void k_bce_pass1(const float* __restrict__ s_map,
                            const float* __restrict__ gtp,
                            const float* __restrict__ maskp,
                            float* __restrict__ negLoss,
                            Acc* __restrict__ acc,
                            float* __restrict__ posPart) {
  extern __shared__ float smem[];           // 2 * 3*CHUNK floats + 16 reduce
  float* sred = smem + 6 * CHUNK;
  float posSum = 0.0f, pcnt = 0.0f, ncnt = 0.0f;
  const int stride = (int)gridDim.x;
#if HAVE_TDM
  const unsigned int ldsBase = __builtin_amdgcn_groupstaticsize();
  int buf = 0;
  {  // prologue: prefetch first chunk into buffer 0
    const int base = (int)blockIdx.x * CHUNK;
    if (threadIdx.x < 32u) {
      tdm_load_1d(s_map + base, ldsBase + 0u,                        CHUNK);
      tdm_load_1d(gtp   + base, ldsBase + (unsigned)(CHUNK * 4),     CHUNK);
      tdm_load_1d(maskp + base, ldsBase + (unsigned)(2 * CHUNK * 4), CHUNK);
    }
  }
#endif
  for (int chunk = (int)blockIdx.x; chunk < NCHUNK; chunk += stride) {
    const int base = chunk * CHUNK;
#if HAVE_TDM
    const int nextChunk = chunk + stride;
    if (threadIdx.x < 32u) {
      if (nextChunk < NCHUNK) {               // prefetch next into buf^1
        const unsigned int nb =
            ldsBase + (unsigned)((buf ^ 1) * (3 * CHUNK * 4));
        const int nbase = nextChunk * CHUNK;
        tdm_load_1d(s_map + nbase, nb + 0u,                        CHUNK);
        tdm_load_1d(gtp   + nbase, nb + (unsigned)(CHUNK * 4),     CHUNK);
        tdm_load_1d(maskp + nbase, nb + (unsigned)(2 * CHUNK * 4), CHUNK);
        __builtin_amdgcn_s_wait_tensorcnt(3); // current buffer's 3 are done
      } else {
        __builtin_amdgcn_s_wait_tensorcnt(0);
      }
    }
    __syncthreads();
    float* cb = smem + buf * (3 * CHUNK);
    const float4* P4 = (const float4*)cb;
    const float4* G4 = (const float4*)(cb + CHUNK);
    const float4* M4 = (const float4*)(cb + 2 * CHUNK);
#else
    const float4* P4 = (const float4*)(s_map + base);
    const float4* G4 = (const float4*)(gtp + base);
    const float4* M4 = (const float4*)(maskp + base);
#endif
    for (int v = (int)threadIdx.x; v < CHUNK / 4; v += (int)blockDim.x) {
      float4 p = P4[v], g = G4[v], m = M4[v];
      float4 nl;
      BCE1(p.x, g.x, m.x, nl.x);
      BCE1(p.y, g.y, m.y, nl.y);
      BCE1(p.z, g.z, m.z, nl.z);
      BCE1(p.w, g.w, m.w, nl.w);
      ((float4*)negLoss)[base / 4 + v] = nl;
    }
#if HAVE_TDM
    __syncthreads();                          // buf will be DMA-overwritten
    buf ^= 1;
#endif
  }
  posSum = blockReduceF(posSum, sred);
  pcnt   = blockReduceF(pcnt, sred);
  ncnt   = blockReduceF(ncnt, sred);
  if (threadIdx.x == 0) {
    posPart[blockIdx.x] = posSum;
    atomicAdd(&acc->posCnt, (unsigned int)(pcnt + 0.5f));
    atomicAdd(&acc->negCnt, (unsigned int)(ncnt + 0.5f));
  }
}

// Radix-select histogram pass `pass` (MSB byte first). L2-resident re-reads.
__global__ void k_hist(const unsigned int* __restrict__ bits,
                       Acc* __restrict__ acc, int pass) {
  __shared__ unsigned int h[256];
  for (int i = (int)threadIdx.x; i < 256; i += (int)blockDim.x) h[i] = 0u;
  __syncthreads();
  const int shift = 24 - 8 * pass;
  const unsigned int msk = (pass == 0) ? 0u : (0xFFFFFFFFu << (shift + 8));
  const unsigned int pre = acc->prefix & msk;
  const uint4* p4 = (const uint4*)bits;
  const int total4 = S_TOT / 4;
#define HPROC(x) if (((x) & msk) == pre) atomicAdd(&h[((x) >> shift) & 255u], 1u)
  for (int i = (int)(blockIdx.x * blockDim.x + threadIdx.x); i < total4;
       i += (int)(gridDim.x * blockDim.x)) {
    uint4 u = p4[i];
    HPROC(u.x); HPROC(u.y); HPROC(u.z); HPROC(u.w);
  }
#undef HPROC
  __syncthreads();
  for (int i = (int)threadIdx.x; i < 256; i += (int)blockDim.x)
    if (h[i]) atomicAdd(&acc->hist[pass][i], h[i]);
}

// Narrow the prefix by one byte (tiny; device-side so no host readback).
__global__ void k_scan(Acc* acc, int pass) {
  if (threadIdx.x != 0) return;
  const int shift = 24 - 8 * pass;
  if (pass == 0) {
    unsigned int k3 = 3u * acc->posCnt;
    unsigned int k = (acc->negCnt < k3) ? acc->negCnt : k3;
    acc->kTotal = k; acc->kNeed = k; acc->prefix = 0u; acc->countGreater = 0u;
  }
  unsigned int need = acc->kNeed, cg = acc->countGreater, pre = acc->prefix;
  int b = 255;
  for (; b >= 0; --b) {
    unsigned int c = acc->hist[pass][b];
    if (need <= c) break;
    need -= c; cg += c;
  }
  if (b < 0) b = 0;
  acc->prefix = pre | ((unsigned int)b << shift);
  acc->kNeed = need;
  acc->countGreater = cg;
}

// Sum of neg losses strictly greater than the k-th largest (bit compare ok:
// all values are non-negative floats).
__global__ void k_sumgt(const unsigned int* __restrict__ bits,
                        const Acc* __restrict__ acc,
                        float* __restrict__ negPart) {
  __shared__ float sred[8];
  const unsigned int thr = acc->prefix;
  float s = 0.0f;
  const uint4* p4 = (const uint4*)bits;
  const int total4 = S_TOT / 4;
  for (int i = (int)(blockIdx.x * blockDim.x + threadIdx.x); i < total4;
       i += (int)(gridDim.x * blockDim.x)) {
    uint4 u = p4[i];
    if (u.x > thr) s += __uint_as_float(u.x);
    if (u.y > thr) s += __uint_as_float(u.y);
    if (u.z > thr) s += __uint_as_float(u.z);
    if (u.w > thr) s += __uint_as_float(u.w);
  }
  s = blockReduceF(s, sred);
  if (threadIdx.x == 0) negPart[blockIdx.x] = s;
}

// gt_e mask at 160x160 (gt[4h,4w] > 0.5) + npix per sample.
__global__ void k_maske(const float* __restrict__ gtp,
                        float* __restrict__ maskE, Acc* __restrict__ acc) {
  __shared__ float sred[8];
  const int idx = (int)(blockIdx.x * blockDim.x + threadIdx.x);  // < NN*EPIX
  const int n = idx / EPIX;
  const int p = idx - n * EPIX;
  const int h = p / WE;
  const int w = p - h * WE;
  float g = gtp[(size_t)n * (HH * WW) + (size_t)(h * 4) * WW + (size_t)(w * 4)];
  float m = (g > 0.5f) ? 1.0f : 0.0f;
  maskE[idx] = m;
  float c = blockReduceF(m, sred);
  if (threadIdx.x == 0)
    atomicAdd(&acc->npix[n], (unsigned int)(c + 0.5f));  // block within one n
}

// One block per (n,c): masked sum / sum-of-squares over 160x160 pixels.
// TDM double-buffered (1 DMA per buffer; tensorcnt<=1 => previous buf ready).
__global__ void k_style(const float* __restrict__ e_map,
                        const float* __restrict__ maskE,
                        float* __restrict__ chSum,
                        float* __restrict__ chSq) {
  extern __shared__ float smem[];           // 2*CHUNK_E floats + 16 reduce
  float* sred = smem + 2 * CHUNK_E;
  const int bid = (int)blockIdx.x;          // n*CC + c
  const int n = bid >> 5;
  const float* e = e_map + (size_t)bid * EPIX;
  const float* m = maskE + (size_t)n * EPIX;
  float s = 0.0f, sq = 0.0f;
#if HAVE_TDM
  const unsigned int ldsBase = __builtin_amdgcn_groupstaticsize();
  int buf = 0;
  if (threadIdx.x < 32u) tdm_load_1d(e, ldsBase, CHUNK_E);
#endif
  for (int ch = 0; ch < NCHE; ++ch) {
#if HAVE_TDM
    if (threadIdx.x < 32u) {
      if (ch + 1 < NCHE) {
        tdm_load_1d(e + (ch + 1) * CHUNK_E,
                    ldsBase + (unsigned)((buf ^ 1) * (CHUNK_E * 4)), CHUNK_E);
        __builtin_amdgcn_s_wait_tensorcnt(1);
      } else {
        __builtin_amdgcn_s_wait_tensorcnt(0);
      }
    }
    __syncthreads();
    const float4* E4 = (const float4*)(smem + buf * CHUNK_E);
#else
    const float4* E4 = (const float4*)(e + ch * CHUNK_E);
#endif
    const float4* M4 = (const float4*)(m + ch * CHUNK_E);
    for (int v = (int)threadIdx.x; v < CHUNK_E / 4; v += (int)blockDim.x) {
      float4 ev = E4[v];
      float4 mv = M4[v];
      s  += ev.x * mv.x + ev.y * mv.y + ev.z * mv.z + ev.w * mv.w;
      sq += ev.x * ev.x * mv.x + ev.y * ev.y * mv.y +
            ev.z * ev.z * mv.z + ev.w * ev.w * mv.w;
    }
#if HAVE_TDM
    __syncthreads();
    buf ^= 1;
#endif
  }
  s  = blockReduceF(s, sred);
  sq = blockReduceF(sq, sred);
  if (threadIdx.x == 0) { chSum[bid] = s; chSq[bid] = sq; }
}

__global__ void k_final(const Acc* __restrict__ acc,
                        const float* __restrict__ posPart,
                        const float* __restrict__ negPart,
                        const float* __restrict__ chSum,
                        const float* __restrict__ chSq,
                        float* __restrict__ out) {
  __shared__ float sred[8];
  float ps = 0.0f;
  for (int i = (int)threadIdx.x; i < NB1; i += 256) ps += posPart[i];
  ps = blockReduceF(ps, sred);
  float ns = 0.0f;
  for (int i = (int)threadIdx.x; i < NB2; i += 256) ns += negPart[i];
  ns = blockReduceF(ns, sred);
  if (threadIdx.x == 0) {
    // ties: kNeed copies of the exact k-th value (bit pattern acc->prefix)
    float tie = (acc->kNeed > 0u)
                    ? __uint_as_float(acc->prefix) * (float)acc->kNeed
                    : 0.0f;
    float negSum = ns + tie;
    float bce = (ps + negSum) /
                ((float)acc->posCnt + (float)acc->kTotal + EPSF);
    float varSum = 0.0f, cnt = 0.0f;
    for (int n = 0; n < NN; ++n) {
      float np = (float)acc->npix[n];
      if (np >= 1.0f) {
        float a = 0.0f;
        for (int c = 0; c < CC; ++c) {
          float sm = chSum[n * CC + c];
          float qq = chSq[n * CC + c];
          a += qq - sm * sm / np;     // == sum(((e-mean)*m)^2) over h,w
        }
        varSum += a / (np + EPSF);
        cnt += 1.0f;
      }
    }
    float style = (cnt > 0.0f) ? (varSum / cnt) : 0.0f;
    out[0] = bce + 0.1f * style;      // BCE_SCALE=1, STYLE_SCALE=0.1
  }
}

// ---------------------------------------------------------------------------
extern "C" void kernel_launch(void* const* d_in, const int* in_sizes, int n_in,
                              void* d_out, int out_size, void* d_ws,
                              size_t ws_size, hipStream_t stream) {
  (void)in_sizes; (void)n_in; (void)out_size; (void)ws_size;
  const float* s_map = (const float*)d_in[0];
  const float* e_map = (const float*)d_in[1];
  const float* gtp   = (const float*)d_in[2];
  const float* maskp = (const float*)d_in[3];
  char* wsb = (char*)d_ws;
  float* negLoss = (float*)(wsb + OFF_NEGLOSS);   // 26.2 MB scratch
  Acc*   acc     = (Acc*)(wsb + OFF_ACC);
  float* posPart = (float*)(wsb + OFF_POSPART);
  float* negPart = (float*)(wsb + OFF_NEGPART);
  float* chSum   = (float*)(wsb + OFF_CHSUM);
  float* chSq    = (float*)(wsb + OFF_CHSQ);
  float* maskE   = (float*)(wsb + OFF_MASKE);
  float* out     = (float*)d_out;

  k_init<<<1, 256, 0, stream>>>(acc);
  k_bce_pass1<<<NB1, 256, (6 * CHUNK + 16) * sizeof(float), stream>>>(
      s_map, gtp, maskp, negLoss, acc, posPart);
  for (int p = 0; p < 4; ++p) {
    k_hist<<<NBH, 256, 0, stream>>>((const unsigned int*)negLoss, acc, p);
    k_scan<<<1, 32, 0, stream>>>(acc, p);
  }
  k_sumgt<<<NB2, 256, 0, stream>>>((const unsigned int*)negLoss, acc, negPart);
  k_maske<<<(NN * EPIX) / 256, 256, 0, stream>>>(gtp, maskE, acc);
  k_style<<<NN * CC, 256, (2 * CHUNK_E + 16) * sizeof(float), stream>>>(
      e_map, maskE, chSum, chSq);
  k_final<<<1, 256, 0, stream>>>(acc, posPart, negPart, chSum, chSq, out);
}